// HybridDiffusion_46377056863083
// MI455X (gfx1250) — compile-verified
//
#include <hip/hip_runtime.h>
#include <hip/hip_bf16.h>
#include <math.h>

// ---------------------------------------------------------------------------
// HybridDiffusion argmax sampler for MI455X (gfx1250).
//
//  - 412 MB streaming read  ->  ~17.7 us roofline at 23.3 TB/s. Memory bound.
//  - No matmul in this op   ->  WMMA not applicable (no v_wmma by design).
//  - Data path: CDNA5 async global->LDS copies (ASYNCcnt-tracked), per-wave
//    pipeline of 4 stages x 32B/lane (two b128 copies per lane per stage),
//    ds_load_b128 consume, LDS tree argmax reduce.
//    In-flight: 4KB/wave, 32KB/block -> ~128KB/WGP with 4 resident blocks,
//    enough to cover ~700ns HBM latency at ~90 GB/s per-WGP share.
//  - log() is monotone and the background prob is a constant -> background
//    argmax of prob/gn == argmax of raw noise. Only 3 candidates per row need
//    the actual log+divide (background winner, MASK column, hit column).
// ---------------------------------------------------------------------------

#define VOCABN   50257
#define MASKIDN  50256
#define SEQN     512
#define NBODY    50256          // columns excluding the MASK column
#define BLOCK    256            // 8 wave32s
#define GPSTAGE  512            // float4 groups per stage (two per thread)
#define STAGES   25             // ceil(12564 / 512); Ng is 12563 or 12564
#define DEPTH    4              // async pipeline depth per wave (slots)

// Low 32 bits of a generic pointer to a __shared__ object are the LDS byte
// offset (the LDS aperture lives entirely in the high 32 address bits).
__device__ __forceinline__ unsigned lds_off_u32(const void* p) {
    return (unsigned)(unsigned long long)p;
}

// One 16B-per-lane async copy: HBM -> LDS, no VGPR destination, ASYNCcnt.
__device__ __forceinline__ void async_copy_b128(unsigned lds_addr, const float* gaddr) {
    asm volatile("global_load_async_to_lds_b128 %0, %1, off"
                 :: "v"(lds_addr), "v"(gaddr)
                 : "memory");
}

__global__ __launch_bounds__(BLOCK)
void hd_argmax_kernel(const int* __restrict__ input_ids,
                      const float* __restrict__ tarr,
                      const float* __restrict__ noise,
                      float* __restrict__ out)
{
    // 8 waves * 4 slots * (32 lanes * 32B) = 32 KB staging ring
    __shared__ __align__(16) float stage[8][DEPTH][256];
    __shared__ float redV[BLOCK];
    __shared__ int   redI[BLOCK];

    const int row = blockIdx.x;
    const int tid = threadIdx.x;
    const int wv  = tid >> 5;          // wave32 id
    const int ln  = tid & 31;          // lane id
    const float* rowp = noise + (size_t)row * VOCABN;

    // Row byte offset = row*201028 == 4*row (mod 16): skip 'a' head elements
    // so the bulk is 16B-aligned float4 groups.
    const int a  = (4 - (row & 3)) & 3;
    const int Ng = (NBODY - a) >> 2;   // aligned float4 groups (12563 or 12564)

    float m    = -1.0f;                // running max noise (noise >= 0)
    int   best = 0;                    // its column index

    if (tid == 0) {                    // head scalars, increasing index order
        for (int c = 0; c < a; ++c) { float v = rowp[c]; if (v > m) { m = v; best = c; } }
    }

    // ---- per-wave async global->LDS pipeline (2 x b128 per lane per stage) ----
    const int laneG0 = wv * 32 + ln;   // 0..255

    auto issue = [&](int s) {
        const int G0 = s * GPSTAGE + 2 * laneG0;   // lane's two consecutive groups
        const unsigned base = lds_off_u32(&stage[wv][s & (DEPTH - 1)][ln * 8]);
        const float* g = rowp + a + 4 * G0;
        if (G0 < Ng)     async_copy_b128(base,      g);
        if (G0 + 1 < Ng) async_copy_b128(base + 16, g + 4);
    };

    for (int s = 0; s < DEPTH; ++s) issue(s);      // prologue: fill the pipe

    for (int s = 0; s < STAGES; ++s) {
        // <=2 ops/stage, in-order retirement: cnt<=6 => stage s is in LDS
        asm volatile("s_wait_asynccnt 0x6" ::: "memory");

        const int G0 = s * GPSTAGE + 2 * laneG0;
        const float* slot = &stage[wv][s & (DEPTH - 1)][ln * 8];

        if (G0 < Ng) {
            const float4 v = *reinterpret_cast<const float4*>(slot);     // ds_load_b128
            float m4 = fmaxf(fmaxf(v.x, v.y), fmaxf(v.z, v.w));
            if (m4 > m) {              // strict > keeps earliest index on ties
                m = m4;
                int j = (v.x == m4) ? 0 : ((v.y == m4) ? 1 : ((v.z == m4) ? 2 : 3));
                best = a + 4 * G0 + j;
            }
        }
        if (G0 + 1 < Ng) {
            const float4 v = *reinterpret_cast<const float4*>(slot + 4); // ds_load_b128
            float m4 = fmaxf(fmaxf(v.x, v.y), fmaxf(v.z, v.w));
            if (m4 > m) {
                m = m4;
                int j = (v.x == m4) ? 0 : ((v.y == m4) ? 1 : ((v.z == m4) ? 2 : 3));
                best = a + 4 * (G0 + 1) + j;
            }
        }

        // make sure the ds_loads above have drained before reusing their slot
        asm volatile("s_wait_dscnt 0x0" ::: "memory");

        if (s + DEPTH < STAGES) issue(s + DEPTH);
    }
    asm volatile("s_wait_asynccnt 0x0" ::: "memory");

    if (tid == 0) {                    // tail scalars (largest indices, strict >)
        for (int c = a + 4 * Ng; c < NBODY; ++c) {
            float v = rowp[c]; if (v > m) { m = v; best = c; }
        }
    }

    // ---- block argmax reduce: value desc, index asc (jnp.argmax semantics) ----
    redV[tid] = m; redI[tid] = best;
    __syncthreads();
    for (int off = BLOCK >> 1; off > 0; off >>= 1) {
        if (tid < off) {
            float vo = redV[tid + off]; int io = redI[tid + off];
            float vm = redV[tid];       int im = redI[tid];
            if (vo > vm || (vo == vm && io < im)) { redV[tid] = vo; redI[tid] = io; }
        }
        __syncthreads();
    }

    if (tid == 0) {
        // Coefficients (GAMMA=1, LOG_B clipped to -20 -> B_EXP = e^-20), f32 math
        // mirroring the reference expression order.
        const float t    = tarr[row / SEQN];
        const float tg   = t;
        const float t1g  = 1.0f - t;
        const float BEXP = 2.0611536224385579e-9f;   // exp(-20)
        const float c    = sqrtf(tg) * sqrtf(t1g) * BEXP;
        float C = (tg + t1g) + (float)(VOCABN - 2) * c;
        C = fmaxf(C, 1.0e-4f);
        const float alpha = (t1g - c) / C;
        const float cC    = c / C;
        const float mC    = tg / C;

        // Background candidate (constant prob cC at the max-noise column).
        int   bi = redI[0];
        float bs = cC / (1.0e-10f - logf(redV[0] + 1.0e-10f));

        // MASK candidate (index 50256 > every background index -> strict >).
        {
            float g = 1.0e-10f - logf(rowp[MASKIDN] + 1.0e-10f);
            float s = mC / g;
            if (s > bs) { bs = s; bi = MASKIDN; }
        }
        // Hit candidate (skipped when the token IS the mask id: the reference's
        // mask branch takes precedence there).
        const int hit = input_ids[row];
        if (hit != MASKIDN) {
            float g = 1.0e-10f - logf(rowp[hit] + 1.0e-10f);
            float s = (alpha + cC) / g;
            if (s > bs || (s == bs && hit < bi)) { bs = s; bi = hit; }
        }
        out[row] = (float)bi;
    }
}

extern "C" void kernel_launch(void* const* d_in, const int* in_sizes, int n_in,
                              void* d_out, int out_size, void* d_ws, size_t ws_size,
                              hipStream_t stream) {
    (void)in_sizes; (void)n_in; (void)d_ws; (void)ws_size;
    const int*   ids   = (const int*)d_in[0];    // input_ids  (B*SEQ)
    const float* tvals = (const float*)d_in[1];  // t          (B)
    const float* noise = (const float*)d_in[2];  // noise      (B*SEQ*VOCAB)
    // out_size == B*SEQ == 2048 rows, one block per row.
    hd_argmax_kernel<<<out_size, BLOCK, 0, stream>>>(ids, tvals, noise, (float*)d_out);
}